// GraphSAGEModel_19284403159491
// MI455X (gfx1250) — compile-verified
//
#include <hip/hip_runtime.h>

#define NN 50000
#define NE 800000

typedef __attribute__((ext_vector_type(2))) float v2f;
typedef __attribute__((ext_vector_type(8))) float v8f;

// ---------------- utility kernels ----------------

__global__ void zero_kernel(float* __restrict__ p, size_t n) {
  size_t i = (size_t)blockIdx.x * blockDim.x + threadIdx.x;
  size_t stride = (size_t)gridDim.x * blockDim.x;
  for (; i < n; i += stride) p[i] = 0.0f;
}

__global__ void deg_count_kernel(const int* __restrict__ dst, float* __restrict__ deg, int E) {
  int e = blockIdx.x * blockDim.x + threadIdx.x;
  if (e < E) atomicAdd(&deg[dst[e]], 1.0f);
}

__global__ void deg_invert_kernel(float* __restrict__ deg, int n) {
  int i = blockIdx.x * blockDim.x + threadIdx.x;
  if (i < n) deg[i] = 1.0f / fmaxf(deg[i], 1.0f);
}

// copy [rows, kin] row-major into zero-padded [rows, kout]
__global__ void pad_kernel(const float* __restrict__ src, float* __restrict__ dstp,
                           int rows, int kin, int kout) {
  size_t total = (size_t)rows * kout;
  size_t i = (size_t)blockIdx.x * blockDim.x + threadIdx.x;
  size_t stride = (size_t)gridDim.x * blockDim.x;
  for (; i < total; i += stride) {
    int r = (int)(i / kout);
    int c = (int)(i % kout);
    dstp[i] = (c < kin) ? src[(size_t)r * kin + c] : 0.0f;
  }
}

// ---------------- edge aggregation (scatter-add) ----------------

template<int D>
__global__ void scatter_add_kernel(const float* __restrict__ feat, const int* __restrict__ src,
                                   const int* __restrict__ dst, float* __restrict__ agg, int E) {
  constexpr int CH = D / 4;  // float4 chunks per edge
  unsigned tid = blockIdx.x * blockDim.x + threadIdx.x;
  unsigned total = (unsigned)E * CH;
  if (tid >= total) return;
  unsigned e = tid / CH;
  unsigned c = (tid - e * CH) * 4u;
  int s = src[e], d = dst[e];
  const float4 v = *reinterpret_cast<const float4*>(feat + (size_t)s * D + c);
  float* op = agg + (size_t)d * D + c;
  atomicAdd(op + 0, v.x);
  atomicAdd(op + 1, v.y);
  atomicAdd(op + 2, v.z);
  atomicAdd(op + 3, v.w);
}

template<int D>
__global__ void scale_rows_kernel(float* __restrict__ agg, const float* __restrict__ dinv, int n) {
  size_t total = (size_t)n * D;
  size_t i = (size_t)blockIdx.x * blockDim.x + threadIdx.x;
  size_t stride = (size_t)gridDim.x * blockDim.x;
  for (; i < total; i += stride) agg[i] *= dinv[i / D];
}

// ---------------- dual GEMM via f32 WMMA ----------------
// out[N,Dout] = A1[N,K] @ W1[Dout,K]^T + A2[N,K] @ W2[Dout,K]^T + bias
// One wave computes a 16 x (16*NT) tile. K stepped by 4 (V_WMMA_F32_16X16X4_F32).

template<int NT>
__global__ void __launch_bounds__(128)
gemm_dual_wmma(const float* __restrict__ A1, const float* __restrict__ W1,
               const float* __restrict__ A2, const float* __restrict__ W2,
               const float* __restrict__ bias, float* __restrict__ out,
               int nRows, int K, int Dout) {
  const int lane = threadIdx.x & 31;
  const int wave = blockIdx.x * (blockDim.x >> 5) + (threadIdx.x >> 5);
  const int colGroups = Dout / (16 * NT);
  const int strip = wave / colGroups;
  if (strip * 16 >= nRows) return;            // wave-uniform exit (EXEC all-1s for WMMA)
  const int cg = wave - strip * colGroups;
  const int rowBase = strip * 16;
  const int colBase = cg * 16 * NT;
  const int lh = lane >> 4;                   // lane half selects K pair
  const int ll = lane & 15;                   // M for A, N for B

  v8f acc[NT];
  const v8f zero8 = {0.f, 0.f, 0.f, 0.f, 0.f, 0.f, 0.f, 0.f};
#pragma unroll
  for (int t = 0; t < NT; ++t) acc[t] = zero8;

  for (int pass = 0; pass < 2; ++pass) {
    const float* __restrict__ A = pass ? A2 : A1;
    const float* __restrict__ W = pass ? W2 : W1;
    const float* arow = A + (size_t)(rowBase + ll) * K;
    for (int k = 0; k < K; k += 4) {
      const int ka = k + lh * 2;
      v2f a;
      a.x = arow[ka];
      a.y = arow[ka + 1];
#pragma unroll
      for (int t = 0; t < NT; ++t) {
        const float* wrow = W + (size_t)(colBase + t * 16 + ll) * K;  // B[k][n] = W[n][k]
        v2f b;
        b.x = wrow[ka];
        b.y = wrow[ka + 1];
        acc[t] = __builtin_amdgcn_wmma_f32_16x16x4_f32(
            false, a, false, b, (short)0, acc[t], false, false);
      }
    }
  }

  // C/D layout: VGPR v holds M = v + lh*8, N = colBase + t*16 + ll
#pragma unroll
  for (int t = 0; t < NT; ++t) {
#pragma unroll
    for (int v = 0; v < 8; ++v) {
      const int m = v + lh * 8;
      const int n = colBase + t * 16 + ll;
      out[(size_t)(rowBase + m) * Dout + n] = acc[t][v] + bias[n];
    }
  }
}

// ---------------- batch norm ----------------

template<int D>
__global__ void bn_stats_kernel(const float* __restrict__ h, float* __restrict__ sum,
                                float* __restrict__ sumsq, int n, int chunks) {
  int tid = blockIdx.x * blockDim.x + threadIdx.x;
  int total = D * chunks;
  if (tid >= total) return;
  int j = tid % D;
  int c = tid / D;
  int per = (n + chunks - 1) / chunks;
  int r0 = c * per;
  int r1 = min(n, r0 + per);
  float s = 0.f, sq = 0.f;
  for (int r = r0; r < r1; ++r) {
    float v = h[(size_t)r * D + j];
    s += v;
    sq += v * v;
  }
  atomicAdd(&sum[j], s);
  atomicAdd(&sumsq[j], sq);
}

template<int D>
__global__ void bn_relu_kernel(float* __restrict__ h, const float* __restrict__ sum,
                               const float* __restrict__ sumsq, const float* __restrict__ g,
                               const float* __restrict__ bt, int n) {
  size_t total = (size_t)n * D;
  size_t i = (size_t)blockIdx.x * blockDim.x + threadIdx.x;
  size_t stride = (size_t)gridDim.x * blockDim.x;
  const float inv_n = 1.0f / (float)n;
  for (; i < total; i += stride) {
    int j = (int)(i % D);
    float m = sum[j] * inv_n;
    float var = fmaxf(sumsq[j] * inv_n - m * m, 0.0f);
    float v = (h[i] - m) * rsqrtf(var + 1e-5f) * g[j] + bt[j];
    h[i] = v > 0.f ? v : 0.f;
  }
}

// ---------------- classifier [N,64] -> [N,2] ----------------

__global__ void classifier_kernel(const float* __restrict__ emb, const float* __restrict__ Wc,
                                  const float* __restrict__ bc, float* __restrict__ logits, int n) {
  int i = blockIdx.x * blockDim.x + threadIdx.x;
  if (i >= n) return;
  const float* e = emb + (size_t)i * 64;
  float s0 = bc[0], s1 = bc[1];
#pragma unroll
  for (int j = 0; j < 64; ++j) {
    float v = e[j];
    s0 = fmaf(v, Wc[j], s0);
    s1 = fmaf(v, Wc[64 + j], s1);
  }
  logits[2 * i + 0] = s0;
  logits[2 * i + 1] = s1;
}

// ---------------- launcher ----------------

extern "C" void kernel_launch(void* const* d_in, const int* in_sizes, int n_in,
                              void* d_out, int out_size, void* d_ws, size_t ws_size,
                              hipStream_t stream) {
  const int N = NN, E = NE;
  const float* x   = (const float*)d_in[0];
  const int*   ei  = (const int*)d_in[1];
  const int* src = ei;          // edge_index[0]
  const int* dst = ei + E;      // edge_index[1]
  const float* Wl1 = (const float*)d_in[2];
  const float* Wr1 = (const float*)d_in[3];
  const float* b1  = (const float*)d_in[4];
  const float* Wl2 = (const float*)d_in[5];
  const float* Wr2 = (const float*)d_in[6];
  const float* b2  = (const float*)d_in[7];
  const float* Wle = (const float*)d_in[8];
  const float* Wre = (const float*)d_in[9];
  const float* be  = (const float*)d_in[10];
  const float* g1  = (const float*)d_in[11];
  const float* bt1 = (const float*)d_in[12];
  const float* g2  = (const float*)d_in[13];
  const float* bt2 = (const float*)d_in[14];
  const float* Wc  = (const float*)d_in[15];
  const float* bc  = (const float*)d_in[16];

  float* outp   = (float*)d_out;
  float* logits = outp;                 // [N,2]
  float* emb    = outp + (size_t)N * 2; // [N,64]

  // workspace layout (floats)
  float* ws = (float*)d_ws;
  size_t o = 0;
  float* deg  = ws + o; o += 50176;             // degree -> 1/max(deg,1)
  float* xpad = ws + o; o += (size_t)N * 176;   // x padded to K=176
  float* agg1 = ws + o; o += (size_t)N * 176;
  float* h1   = ws + o; o += (size_t)N * 256;
  float* h2   = ws + o; o += (size_t)N * 128;
  float* sum  = ws + o; o += 512;
  float* sq   = ws + o; o += 512;
  float* wl1p = ws + o; o += 256 * 176;
  float* wr1p = ws + o; o += 256 * 176;
  float* agg2 = xpad;  // reuse: xpad/agg1 dead after layer-1 GEMM (needs N*256 <= 2*N*176)
  float* agg3 = xpad;  // reuse: agg2 dead after layer-2 GEMM

  const int TB = 256;
  auto blk = [](size_t n, int tb) { return (unsigned)((n + (size_t)tb - 1) / tb); };

  // degrees (shared by all layers)
  zero_kernel<<<256, TB, 0, stream>>>(deg, (size_t)N);
  deg_count_kernel<<<blk(E, TB), TB, 0, stream>>>(dst, deg, E);
  deg_invert_kernel<<<blk(N, TB), TB, 0, stream>>>(deg, N);

  // pad inputs to K=176
  pad_kernel<<<4096, TB, 0, stream>>>(x, xpad, N, 166, 176);
  pad_kernel<<<64, TB, 0, stream>>>(Wl1, wl1p, 256, 166, 176);
  pad_kernel<<<64, TB, 0, stream>>>(Wr1, wr1p, 256, 166, 176);

  // ---- layer 1: 176 -> 256, BN + ReLU ----
  zero_kernel<<<4096, TB, 0, stream>>>(agg1, (size_t)N * 176);
  scatter_add_kernel<176><<<blk((size_t)E * 44, TB), TB, 0, stream>>>(xpad, src, dst, agg1, E);
  scale_rows_kernel<176><<<4096, TB, 0, stream>>>(agg1, deg, N);
  {
    int waves = (N / 16) * (256 / 64);
    gemm_dual_wmma<4><<<(waves + 3) / 4, 128, 0, stream>>>(agg1, wl1p, xpad, wr1p, b1, h1, N, 176, 256);
  }
  zero_kernel<<<4, TB, 0, stream>>>(sum, 1024);  // sum + sq contiguous
  bn_stats_kernel<256><<<blk(256 * 200, TB), TB, 0, stream>>>(h1, sum, sq, N, 200);
  bn_relu_kernel<256><<<4096, TB, 0, stream>>>(h1, sum, sq, g1, bt1, N);

  // ---- layer 2: 256 -> 128, BN + ReLU ----
  zero_kernel<<<4096, TB, 0, stream>>>(agg2, (size_t)N * 256);
  scatter_add_kernel<256><<<blk((size_t)E * 64, TB), TB, 0, stream>>>(h1, src, dst, agg2, E);
  scale_rows_kernel<256><<<4096, TB, 0, stream>>>(agg2, deg, N);
  {
    int waves = (N / 16) * (128 / 64);
    gemm_dual_wmma<4><<<(waves + 3) / 4, 128, 0, stream>>>(agg2, Wl2, h1, Wr2, b2, h2, N, 256, 128);
  }
  zero_kernel<<<4, TB, 0, stream>>>(sum, 1024);
  bn_stats_kernel<128><<<blk(128 * 200, TB), TB, 0, stream>>>(h2, sum, sq, N, 200);
  bn_relu_kernel<128><<<4096, TB, 0, stream>>>(h2, sum, sq, g2, bt2, N);

  // ---- layer 3: 128 -> 64 (embeddings, no BN/ReLU) ----
  zero_kernel<<<4096, TB, 0, stream>>>(agg3, (size_t)N * 128);
  scatter_add_kernel<128><<<blk((size_t)E * 32, TB), TB, 0, stream>>>(h2, src, dst, agg3, E);
  scale_rows_kernel<128><<<4096, TB, 0, stream>>>(agg3, deg, N);
  {
    int waves = (N / 16) * (64 / 64);
    gemm_dual_wmma<4><<<(waves + 3) / 4, 128, 0, stream>>>(agg3, Wle, h2, Wre, be, emb, N, 128, 64);
  }

  // classifier
  classifier_kernel<<<blk(N, TB), TB, 0, stream>>>(emb, Wc, bc, logits, N);
}